// OrthogonalAddTSU_3728031613721
// MI455X (gfx1250) — compile-verified
//
#include <hip/hip_runtime.h>
#include <stdint.h>

// Problem sizes (match reference).
#define BDIM   64
#define HDIM   1024
#define SDIM   512
#define EPS    1e-8f

// Tiling: TDM copies (HDIM rows) x (TC time-steps) tiles into LDS.
// TDM pad feature: +1 dword after every 4 dwords -> LDS row stride 5 dwords.
// Bank for (j, tt) = (5*j + tt) mod 64; across lanes j = lane + 32k this is
// 5*lane mod 64 -> all 32 distinct (gcd(5,64)=1): conflict-free scalar reads.
#define TC      4
#define ROWPAD  5
#define NTILES  (SDIM / TC)          // 128
#define NBUF    3
#define BUFDW   (HDIM * ROWPAD)      // 5120 dwords = 20 KiB per buffer
#define KPL     (HDIM / 32)          // 32 h-elements per lane
#define PPL     (KPL / 2)            // 16 float2 pairs per lane

typedef unsigned int v4u __attribute__((ext_vector_type(4)));
typedef int          v4i __attribute__((ext_vector_type(4)));
typedef int          v8i __attribute__((ext_vector_type(8)));
typedef float        v2f __attribute__((ext_vector_type(2)));

// Native vector FMA: backend selects v_pk_fma_f32 (packed-FP32 VOP3P) with
// full regalloc/scheduler integration (no inline-asm pairing movs).
__device__ __forceinline__ v2f vfma(v2f a, v2f b, v2f c) {
  return __builtin_elementwise_fma(a, b, c);
}
// Clamp to [-1,1]; per-half min(max()) folds to v_med3_f32 / v_max+v_min.
__device__ __forceinline__ v2f vclamp11(v2f x) {
  const v2f lo = {-1.0f, -1.0f};
  const v2f hi = { 1.0f,  1.0f};
  return __builtin_elementwise_min(__builtin_elementwise_max(x, lo), hi);
}

// ---- wave32 xor-reduction via ds_swizzle (all lanes end with the full sum) ----
template <int MASK>
__device__ __forceinline__ float swz_add(float x) {
  int y = __builtin_amdgcn_ds_swizzle(__float_as_int(x), (MASK << 10) | 0x1f);
  return x + __int_as_float(y);
}
__device__ __forceinline__ float wave_sum32(float x) {
  x = swz_add<16>(x);
  x = swz_add<8>(x);
  x = swz_add<4>(x);
  x = swz_add<2>(x);
  x = swz_add<1>(x);
  return x;
}

// ---- Tensor Data Mover: DMA one (HDIM x TC) tile, global -> LDS ----
// D# per CDNA5 ISA. 2D tensor: groups 2/3 zero.
// 6-arg builtin form (clang-23 / therock-10.0):
//   (uint32x4 g0, int32x8 g1, int32x4 g2, int32x4 g3, int32x8 g4, i32 cpol)
__device__ __forceinline__ void tdm_tile_load(unsigned lds_byte_off, const float* gptr) {
  uint64_t ga = (uint64_t)(uintptr_t)gptr;

  v4u g0;
  g0.x = 1u;                                                  // count=1, user descriptor
  g0.y = lds_byte_off;                                        // lds_addr (bytes)
  g0.z = (unsigned)(ga & 0xffffffffull);                      // global_addr[31:0]
  g0.w = (unsigned)((ga >> 32) & 0x1ffffffull) | (2u << 30);  // addr[56:32] | type=2

  v8i g1;
  g1[0] = (int)((2u << 16)      // data_size = 4 bytes
              | (1u << 20)      // pad_enable
              | (1u << 22)      // pad_interval: code 1 = every 4 dwords
              | (0u << 25));    // pad_amount:   code 0 = 1 dword
  g1[1] = (int)(((unsigned)SDIM & 0xffffu) << 16);            // tensor_dim0[15:0]=512
  g1[2] = (int)((((unsigned)SDIM >> 16) & 0xffffu)
              | (((unsigned)HDIM & 0xffffu) << 16));          // td0 hi | tensor_dim1 lo = 1024
  g1[3] = (int)((((unsigned)HDIM >> 16) & 0xffffu)
              | ((unsigned)TC << 16));                        // td1 hi | tile_dim0 = 4
  g1[4] = (int)((unsigned)HDIM & 0xffffu);                    // tile_dim1=1024, tile_dim2=0
  g1[5] = (int)SDIM;                                          // tensor_dim0_stride = 512 elems
  g1[6] = 0;                                                  // stride hi | dim1_stride lo
  g1[7] = 0;

  v4i z4 = {0, 0, 0, 0};
  v8i z8 = {0, 0, 0, 0, 0, 0, 0, 0};
  __builtin_amdgcn_tensor_load_to_lds(g0, g1, z4, z4, z8, 0);
}

extern "C" __global__ __launch_bounds__(32, 1)
void orth_scan_kernel(const float* __restrict__ tree,
                      const float* __restrict__ seq,
                      const float* __restrict__ mask,
                      float* __restrict__ out)
{
  extern __shared__ float lds[];   // NBUF * BUFDW floats (61440 bytes)

  const int b    = blockIdx.x;
  const int lane = threadIdx.x;
  const float* __restrict__ seq_b  = seq  + (size_t)b * HDIM * SDIM;
  const float* __restrict__ mask_b = mask + (size_t)b * SDIM;

  // h carry lives entirely in registers as float2 pairs:
  // pair p holds j = lane + 32*(2p) and j = lane + 32*(2p+1).
  v2f h2[PPL];
  float hh_p;                      // per-lane partial of ||h||^2 (kept incrementally)
  {
    v2f acc = {0.f, 0.f};
#pragma unroll
    for (int p = 0; p < PPL; ++p) {
      h2[p].x = tree[(size_t)b * HDIM + lane + 32 * (2 * p)];
      h2[p].y = tree[(size_t)b * HDIM + lane + 32 * (2 * p + 1)];
      acc = vfma(h2[p], h2[p], acc);
    }
    hh_p = acc.x + acc.y;
  }

  const unsigned lds_base = (unsigned)(uintptr_t)(void*)&lds[0];

  // Pipeline: prefetch tiles 0 and 1.
  tdm_tile_load(lds_base + 0u * BUFDW * 4u, seq_b + 0 * TC);
  tdm_tile_load(lds_base + 1u * BUFDW * 4u, seq_b + 1 * TC);

  auto process = [&](int tile, int buf) {
    const int base = buf * BUFDW;
#pragma unroll
    for (int tt = 0; tt < TC; ++tt) {
      const float m = mask_b[tile * TC + tt];       // uniform -> scalar load
      v2f sv2[PPL];
      v2f dot2 = {0.f, 0.f}, ss2 = {0.f, 0.f};
#pragma unroll
      for (int p = 0; p < PPL; ++p) {
        const int j0 = lane + 32 * (2 * p);
        sv2[p].x = lds[base + j0 * ROWPAD + tt];              // conflict-free ds loads
        sv2[p].y = lds[base + (j0 + 32) * ROWPAD + tt];
        dot2 = vfma(h2[p], sv2[p], dot2);
        ss2  = vfma(sv2[p], sv2[p], ss2);
      }
      const float dot = wave_sum32(dot2.x + dot2.y);
      const float ss  = wave_sum32(ss2.x + ss2.y);
      const float hh  = wave_sum32(hh_p);
      const float denom = fmaxf(sqrtf(hh) * sqrtf(ss), EPS);
      const float cosv  = dot / denom;
      const v2f m2  = {m, m};
      const v2f nc2 = {-cosv, -cosv};
      v2f nhh2 = {0.f, 0.f};
#pragma unroll
      for (int p = 0; p < PPL; ++p) {
        v2f u  = vfma(nc2, h2[p], sv2[p]);                    // s - h*cos
        v2f hn = vfma(m2, u, h2[p]);                          // h + m*(s - h*cos)
        hn = vclamp11(hn);                                    // med3 / min+max per half
        h2[p] = hn;
        nhh2 = vfma(hn, hn, nhh2);
      }
      hh_p = nhh2.x + nhh2.y;
    }
  };

  int tile = 0, buf = 0;
  for (; tile < NTILES - 2; ++tile) {
    int pbuf = buf + 2; if (pbuf >= NBUF) pbuf -= NBUF;
    tdm_tile_load(lds_base + (unsigned)pbuf * BUFDW * 4u,
                  seq_b + (size_t)(tile + 2) * TC);
    __builtin_amdgcn_s_wait_tensorcnt(2);   // oldest (current tile) complete
    process(tile, buf);
    if (++buf == NBUF) buf = 0;
  }
  __builtin_amdgcn_s_wait_tensorcnt(1);
  process(tile, buf);
  ++tile; if (++buf == NBUF) buf = 0;
  __builtin_amdgcn_s_wait_tensorcnt(0);
  process(tile, buf);

#pragma unroll
  for (int p = 0; p < PPL; ++p) {
    out[(size_t)b * HDIM + lane + 32 * (2 * p)]     = h2[p].x;
    out[(size_t)b * HDIM + lane + 32 * (2 * p + 1)] = h2[p].y;
  }
}

extern "C" void kernel_launch(void* const* d_in, const int* in_sizes, int n_in,
                              void* d_out, int out_size, void* d_ws, size_t ws_size,
                              hipStream_t stream) {
  (void)in_sizes; (void)n_in; (void)out_size; (void)d_ws; (void)ws_size;
  const float* tree = (const float*)d_in[0];   // (B,H)
  const float* seq  = (const float*)d_in[1];   // (B,H,S)
  const float* mask = (const float*)d_in[2];   // (B,S)
  float* out = (float*)d_out;                  // (B,H)

  const size_t shmem = (size_t)NBUF * BUFDW * sizeof(float);  // 61440 bytes
  orth_scan_kernel<<<dim3(BDIM), dim3(32), shmem, stream>>>(tree, seq, mask, out);
}